// CliffordBatchNormMV_13950053777613
// MI455X (gfx1250) — compile-verified
//
#include <hip/hip_runtime.h>

typedef __attribute__((ext_vector_type(2))) float v2f;
typedef __attribute__((ext_vector_type(4))) float v4f;
typedef __attribute__((ext_vector_type(8))) float v8f;

#define EPSF 1e-5f

// Fixed problem dims from the reference
constexpr int B_DIM = 4096;
constexpr int NB    = 64;
constexpr int MV    = 256;               // components per multivector
constexpr int NMV   = B_DIM * NB;        // 262144 multivectors
constexpr int NGRD  = 9;
constexpr int CH    = NGRD * NB;         // 576 (g,nb) channels

// Phase-1 geometry: 512 blocks x (8 waves * 4 passes * 16 mv) = 262144 mvs
constexpr int K1_BLOCKS = 512;
constexpr int K1_MV_PER_BLOCK = NMV / K1_BLOCKS;       // 512
constexpr int K1_PASSES = K1_MV_PER_BLOCK / (8 * 16);  // 4

// Workspace layout (in floats)
constexpr size_t INV_STRIDE = 12;                       // 9 grades + pad
constexpr size_t INV_FLOATS = (size_t)NMV * INV_STRIDE; // 3,145,728
constexpr size_t PART_FLOATS = (size_t)K1_BLOCKS * 2 * CH; // 589,824

// ---------------------------------------------------------------------------
// Kernel 1: per-multivector grade sums via V_WMMA_F32_16X16X4_F32
//   ssum[16 mv][16 grades] = Xsq[16][256] @ M[256][16], M[k][g] = (popc(k)==g)
// A layout (f32 16x4): lanes 0-15 = rows(M=mv), VGPR0/1 = K0/K1; lanes 16-31
// hold K2/K3 of the same rows  -> one float2 load at (row, c4 + 2*hi).
// B layout (f32 4x16): VGPR0/1 hold K rows striped across lanes (N=lane&15),
// lower lanes K0/K1, upper lanes K2/K3 -> indicator built from popcounts.
// D layout: lane l, VGPR r  ->  mv = r + 8*(l>>4), grade = l&15.
// ---------------------------------------------------------------------------
__global__ __launch_bounds__(256) void k1_stats(const float* __restrict__ x,
                                                float* __restrict__ wsInv,
                                                float* __restrict__ wsPart)
{
    __shared__ float lsum[8][CH];   // per-wave partials: deterministic, race-free
    __shared__ float lsq [8][CH];

    const int tid  = threadIdx.x;
    const int wave = tid >> 5;
    const int lane = tid & 31;
    const int n    = lane & 15;     // matrix row/col index
    const int hi   = lane >> 4;     // 0: K{0,1}, 1: K{2,3}
    const int nmhi = n - hi;        // indicator test value (B-matrix build)

    for (int i = lane; i < CH; i += 32) { lsum[wave][i] = 0.f; lsq[wave][i] = 0.f; }
    __syncthreads();

    for (int pass = 0; pass < K1_PASSES; ++pass) {
        const int mv_base = blockIdx.x * K1_MV_PER_BLOCK + (pass * 8 + wave) * 16;
        const float* xr = x + (size_t)(mv_base + n) * MV + 2 * hi;

        v8f acc0 = {0,0,0,0,0,0,0,0};
        v8f acc1 = {0,0,0,0,0,0,0,0};
#pragma unroll
        for (int c8 = 0; c8 < MV; c8 += 8) {
            {
                v2f xv = *(const v2f*)(xr + c8);
                v2f a; a.x = xv.x * xv.x; a.y = xv.y * xv.y;
                const int pcc = __builtin_popcount(c8);       // compile-time
                v2f b; b.x = (pcc     == nmhi) ? 1.f : 0.f;   // popc(K)==N
                       b.y = (pcc + 1 == nmhi) ? 1.f : 0.f;
                acc0 = __builtin_amdgcn_wmma_f32_16x16x4_f32(
                    false, a, false, b, (short)0, acc0, false, false);
            }
            {
                v2f xv = *(const v2f*)(xr + c8 + 4);
                v2f a; a.x = xv.x * xv.x; a.y = xv.y * xv.y;
                const int pcc = __builtin_popcount(c8 + 4);
                v2f b; b.x = (pcc     == nmhi) ? 1.f : 0.f;
                       b.y = (pcc + 1 == nmhi) ? 1.f : 0.f;
                acc1 = __builtin_amdgcn_wmma_f32_16x16x4_f32(
                    false, a, false, b, (short)0, acc1, false, false);
            }
        }
        v8f acc = acc0 + acc1;

        if (n < NGRD) {                      // lane's grade column is valid
            const int mvr0 = mv_base + 8 * hi;
#pragma unroll
            for (int r = 0; r < 8; ++r) {
                float s2  = acc[r] + EPSF;
                float inv = rsqrtf(s2);
                float nf  = s2 * inv;        // sqrt(s2)
                wsInv[(size_t)(mvr0 + r) * INV_STRIDE + n] = inv;
                // distinct (grade,nb) per active lane & per r -> no races
                const int idx = n * NB + ((mvr0 + r) & (NB - 1));
                lsum[wave][idx] += nf;
                lsq [wave][idx] += nf * nf;
            }
        }
    }

    __syncthreads();
    float* part = wsPart + (size_t)blockIdx.x * (2 * CH);
    for (int i = tid; i < CH; i += 256) {
        float s = 0.f, q = 0.f;
#pragma unroll
        for (int w = 0; w < 8; ++w) { s += lsum[w][i]; q += lsq[w][i]; }
        part[i]      = s;
        part[CH + i] = q;
    }
}

// ---------------------------------------------------------------------------
// Kernel 2: reduce 512 block-partials per (g,nb), emit (A,B) coefficient table
// ---------------------------------------------------------------------------
__global__ __launch_bounds__(256) void k2_finalize(const float* __restrict__ wsPart,
                                                   const float* __restrict__ gamma,
                                                   const float* __restrict__ beta,
                                                   const float* __restrict__ gscale,
                                                   v2f* __restrict__ wsAB)
{
    __shared__ float rs[256], rq[256];
    const int ch  = blockIdx.x;          // g*64 + nb
    const int tid = threadIdx.x;

    float s = 0.f, q = 0.f;
    for (int i = tid; i < K1_BLOCKS; i += 256) {
        s += wsPart[(size_t)i * (2 * CH) + ch];
        q += wsPart[(size_t)i * (2 * CH) + CH + ch];
    }
    rs[tid] = s; rq[tid] = q;
    __syncthreads();
    for (int off = 128; off > 0; off >>= 1) {
        if (tid < off) { rs[tid] += rs[tid + off]; rq[tid] += rq[tid + off]; }
        __syncthreads();
    }
    if (tid == 0) {
        const float mean = rs[0] * (1.f / B_DIM);
        const float var  = rq[0] * (1.f / B_DIM) - mean * mean;
        const float rstd = rsqrtf(var + EPSF);
        const int   g    = ch >> 6;
        const float gm   = gamma[ch];
        const float gs   = gscale[g];
        v2f ab;
        ab.x = rstd * gm * gs;                       // A
        ab.y = (beta[ch] - mean * rstd * gm) * gs;   // B
        wsAB[ch] = ab;
    }
}

// ---------------------------------------------------------------------------
// Kernel 3: out[c] = x[c] * (A[g,nb] + B[g,nb]*invnf[mv,g]), g = popc(c)
// 8 lanes per mv; grade = popc(lane8) + popc(j) + popc(e): only popc(lane8)
// is runtime -> S[0..5] stays in named registers, fully coalesced b128 I/O.
// ---------------------------------------------------------------------------
__global__ __launch_bounds__(256) void k3_apply(const float* __restrict__ x,
                                                const float* __restrict__ wsInv,
                                                const v2f* __restrict__ wsAB,
                                                float* __restrict__ out)
{
    __shared__ v2f sAB[CH];
    const int tid = threadIdx.x;
    for (int i = tid; i < CH; i += 256) sAB[i] = wsAB[i];
    __syncthreads();

    const int grp = tid >> 3;
    const int l8  = tid & 7;
    const int p   = __popc(l8);          // 0..3

    for (int pass = 0; pass < 8; ++pass) {
        const int mv = blockIdx.x * 256 + pass * 32 + grp;
        const int nb = mv & (NB - 1);

        const float* ivp = wsInv + (size_t)mv * INV_STRIDE + p;
        float S[6];
#pragma unroll
        for (int k = 0; k < 6; ++k) {    // grades p..p+5 (<=8)
            float iv = ivp[k];
            v2f  ab  = sAB[(p + k) * NB + nb];
            S[k] = ab.x + ab.y * iv;
        }

        const float* xr  = x   + (size_t)mv * MV + l8 * 4;
        float*       orr = out + (size_t)mv * MV + l8 * 4;
#pragma unroll
        for (int j = 0; j < 8; ++j) {
            v4f xv = *(const v4f*)(xr + j * 32);
            const int pj = __builtin_popcount(j);    // compile-time
            v4f ov;
            ov.x = xv.x * S[pj + 0];   // e=0: popc 0
            ov.y = xv.y * S[pj + 1];   // e=1: popc 1
            ov.z = xv.z * S[pj + 1];   // e=2: popc 1
            ov.w = xv.w * S[pj + 2];   // e=3: popc 2
            *(v4f*)(orr + j * 32) = ov;
        }
    }
}

// ---------------------------------------------------------------------------
extern "C" void kernel_launch(void* const* d_in, const int* in_sizes, int n_in,
                              void* d_out, int out_size, void* d_ws, size_t ws_size,
                              hipStream_t stream)
{
    const float* x      = (const float*)d_in[0];
    const float* gamma  = (const float*)d_in[1];
    const float* beta   = (const float*)d_in[2];
    const float* gscale = (const float*)d_in[3];
    float* out = (float*)d_out;

    float* wsInv  = (float*)d_ws;                 // NMV * 12 floats (12.6 MB)
    float* wsPart = wsInv + INV_FLOATS;           // 512 * 1152 floats (2.4 MB)
    v2f*   wsAB   = (v2f*)(wsPart + PART_FLOATS); // 576 float2

    k1_stats   <<<K1_BLOCKS, 256, 0, stream>>>(x, wsInv, wsPart);
    k2_finalize<<<CH,        256, 0, stream>>>(wsPart, gamma, beta, gscale, wsAB);
    k3_apply   <<<NMV / 256, 256, 0, stream>>>(x, wsInv, wsAB, out);
}